// RCNNDecoder_15719580304002
// MI455X (gfx1250) — compile-verified
//
#include <hip/hip_runtime.h>
#include <stdint.h>

#define BATCH 8
#define NROI 8192
#define NCLS 80
#define NC (NROI * NCLS)          // 655360 per image
#define TOPK 1000
#define KPAD 1024
#define NBINS 65536
#define CAND_MAX 8192
#define NMS_THR_F 0.5f
#define SCORE_THR_F 0.05f
#define MAX_RATIO_F 4.1351665565490723f   // log(1000/16)
#define CLS_OFFSET_F 10000.0f

typedef unsigned int v4u __attribute__((ext_vector_type(4)));
typedef int v8i __attribute__((ext_vector_type(8)));
typedef int v4i __attribute__((ext_vector_type(4)));

#if defined(__AMDGCN__) && __has_builtin(__builtin_amdgcn_tensor_load_to_lds)
#define HAVE_TDM 1
#else
#define HAVE_TDM 0
#endif

// ---------------------------------------------------------------- utilities
static __device__ __forceinline__ bool wave_any32(bool p) {
#if defined(__AMDGCN__) && __has_builtin(__builtin_amdgcn_ballot_w32)
  return __builtin_amdgcn_ballot_w32(p) != 0u;
#else
  return __any((int)p) != 0;
#endif
}

#if HAVE_TDM
// 1D TDM copy: nelem_tensor valid elements (reads past it return ZERO -> free
// zero-padding up to nelem_tile), data_size_log2 in {0,1,2,3} bytes.
// D# layout per CDNA5 ISA ch.8 (group0: count/lds/global/type, group1: dims).
static __device__ __forceinline__ void tdm_load_1d(const void* gsrc, void* ldst,
                                                   uint32_t nelem_tensor,
                                                   uint32_t nelem_tile,
                                                   uint32_t data_size_log2) {
  uint64_t ga = (uint64_t)(uintptr_t)gsrc;
  uint32_t ldsoff = (uint32_t)(uintptr_t)ldst;   // flat addr low 32 = LDS offset
  v4u g0;
  g0.x = 1u;                                                 // count=1, user D#
  g0.y = ldsoff;                                             // lds_addr
  g0.z = (uint32_t)ga;                                       // global_addr[31:0]
  g0.w = ((uint32_t)(ga >> 32) & 0x01FFFFFFu) | (2u << 30);  // addr[56:32] | type=2
  v8i g1;
  g1[0] = (int)(data_size_log2 << 16);                       // data_size; flags=0
  g1[1] = (int)((nelem_tensor & 0xFFFFu) << 16);             // tensor_dim0[15:0]
  g1[2] = (int)((nelem_tensor >> 16) | (1u << 16));          // dim0[31:16] | dim1=1
  g1[3] = (int)(nelem_tile << 16);                           // tile_dim0
  g1[4] = 1;                                                 // tile_dim1=1, tile_dim2=0
  g1[5] = (int)nelem_tensor;                                 // dim0 stride (single row)
  g1[6] = 0;
  g1[7] = 0;
  v4i gz = {0, 0, 0, 0};
#if defined(__clang_major__) && (__clang_major__ >= 23)
  v8i gz8 = {0, 0, 0, 0, 0, 0, 0, 0};
  __builtin_amdgcn_tensor_load_to_lds(g0, g1, gz, gz, gz8, 0);
#else
  __builtin_amdgcn_tensor_load_to_lds(g0, g1, gz, gz, 0);
#endif
  __builtin_amdgcn_s_wait_tensorcnt(0);
}
#endif

// ---------------------------------------------------------------- kernel 6: IoU suppression bitmask (TDM-fed LDS)
// (defined first so the disasm snippet shows the tensor_load_to_lds sequence)
__global__ __launch_bounds__(256) void k_iou(const float* __restrict__ top_adj,
                                             uint32_t* __restrict__ supmask) {
  __shared__ __align__(16) float ldsAdj[KPAD * 4];   // 16 KB adjusted-box table
  int img = blockIdx.x >> 7;                         // 128 row-tile blocks per image
  int rowTile = blockIdx.x & 127;
  int tid = threadIdx.x;
  const float* src = top_adj + (size_t)img * KPAD * 4;

#if HAVE_TDM
  if (tid == 0) tdm_load_1d(src, (void*)ldsAdj, KPAD * 4, KPAD * 4, 2u);
#else
  for (int idx = tid; idx < KPAD * 4; idx += 256) ldsAdj[idx] = src[idx];
#endif
  __syncthreads();

  int i = rowTile * 8 + (tid >> 5);
  int w = tid & 31;
  const float4* lb = (const float4*)ldsAdj;
  float4 bi = lb[i];
  float ai = (bi.z - bi.x) * (bi.w - bi.y);
  uint32_t word = 0u;
  int j0 = w * 32;
#pragma unroll
  for (int b = 0; b < 32; ++b) {
    float4 bj = lb[j0 + b];
    float aj = (bj.z - bj.x) * (bj.w - bj.y);
    float iw = fmaxf(fminf(bi.z, bj.z) - fmaxf(bi.x, bj.x), 0.f);
    float ih = fmaxf(fminf(bi.w, bj.w) - fmaxf(bi.y, bj.y), 0.f);
    float inter = iw * ih;
    float iou = inter / (ai + aj - inter + 1e-9f);
    word |= (iou > NMS_THR_F) ? (1u << b) : 0u;
  }
  supmask[((size_t)img * KPAD + (size_t)i) * 32 + w] = word;
}

// ---------------------------------------------------------------- kernel 7: sequential NMS scan (wave32 ballot) + output
__global__ __launch_bounds__(1024) void k_nms(const uint32_t* __restrict__ supmask,
                                              const uint32_t* __restrict__ top_valid,
                                              const float* __restrict__ top_box,
                                              const float* __restrict__ top_score,
                                              const float* __restrict__ top_cls,
                                              float* __restrict__ out) {
  __shared__ __align__(16) uint32_t sup[TOPK * 32];  // 125 KB suppression matrix
  __shared__ uint32_t keepw[32];
  __shared__ uint32_t validw[32];
  int img = blockIdx.x;
  int tid = threadIdx.x;
  const uint32_t* sm = supmask + (size_t)img * KPAD * 32;

#if HAVE_TDM
  if (tid == 0) tdm_load_1d(sm, (void*)sup, TOPK * 32, TOPK * 32, 2u);
#else
  for (int i = tid; i < TOPK * 32; i += 1024) sup[i] = sm[i];
#endif
  if (tid < 32) validw[tid] = 0u;
  __syncthreads();
  if (tid < TOPK && top_valid[(size_t)img * KPAD + tid])
    atomicOr(&validw[tid >> 5], 1u << (tid & 31));
  __syncthreads();

  if (tid < 32) {                     // wave 0: lane l owns keep-word l (wave32 == 32 words)
    uint32_t kw = 0u;
    uint32_t vw = validw[tid];
    for (int i = 0; i < TOPK; ++i) {
      uint32_t rw = sup[i * 32 + tid];
      bool hit = wave_any32((kw & rw) != 0u);
      if (tid == (i >> 5)) {
        uint32_t bit = (uint32_t)(i & 31);
        if (((vw >> bit) & 1u) && !hit) kw |= (1u << bit);
      }
    }
    keepw[tid] = kw;
  }
  __syncthreads();

  if (tid < TOPK) {
    size_t slot = (size_t)img * KPAD + tid;
    float kf = ((keepw[tid >> 5] >> (tid & 31)) & 1u) ? 1.0f : 0.0f;
    float* o = out + ((size_t)img * TOPK + tid) * 7;
    o[0] = top_box[slot * 4 + 0] * kf;
    o[1] = top_box[slot * 4 + 1] * kf;
    o[2] = top_box[slot * 4 + 2] * kf;
    o[3] = top_box[slot * 4 + 3] * kf;
    o[4] = top_score[slot] * kf;
    o[5] = top_cls[slot] * kf;
    o[6] = kf;
  }
}

// ---------------------------------------------------------------- kernel 5: per-image bitonic sort + top-K gather
__global__ __launch_bounds__(1024) void k_sort(const uint64_t* __restrict__ cand,
                                               const uint32_t* __restrict__ cnt,
                                               const float* __restrict__ boxes,
                                               float* __restrict__ top_adj,
                                               float* __restrict__ top_box,
                                               float* __restrict__ top_score,
                                               float* __restrict__ top_cls,
                                               uint32_t* __restrict__ top_valid) {
  __shared__ __align__(16) uint64_t keys[CAND_MAX];  // 64 KB
  int img = blockIdx.x;
  int tid = threadIdx.x;
  uint32_t m = cnt[img];
  if (m > CAND_MAX) m = CAND_MAX;
  const uint64_t* src = cand + (size_t)img * CAND_MAX;

#if HAVE_TDM
  // TDM load of m keys; reads past tensor_dim return zero => free zero-padding.
  if (tid == 0) tdm_load_1d(src, (void*)keys, m, CAND_MAX, 3u);
#else
  for (int i = tid; i < CAND_MAX; i += 1024) keys[i] = (i < (int)m) ? src[i] : 0ull;
#endif
  __syncthreads();

  // bitonic sort, descending
  for (unsigned k = 2; k <= CAND_MAX; k <<= 1) {
    for (unsigned j = k >> 1; j > 0; j >>= 1) {
      for (unsigned i = tid; i < CAND_MAX; i += 1024) {
        unsigned ixj = i ^ j;
        if (ixj > i) {
          uint64_t a = keys[i], b = keys[ixj];
          bool sw = ((i & k) == 0) ? (a < b) : (a > b);
          if (sw) { keys[i] = b; keys[ixj] = a; }
        }
      }
      __syncthreads();
    }
  }
  // gather: one slot per thread
  int t = tid;
  size_t slot = (size_t)img * KPAD + t;
  float bx0 = 0.f, bx1 = 0.f, bx2 = 0.f, bx3 = 0.f, sc = 0.f, cl = 0.f;
  float a0 = 0.f, a1 = 0.f, a2 = 0.f, a3 = 0.f;
  uint32_t v = 0u;
  if (t < TOPK) {
    uint64_t key = keys[t];
    sc = __uint_as_float((uint32_t)(key >> 32));
    if (sc > SCORE_THR_F) {
      v = 1u;
      uint32_t f = ~(uint32_t)key;
      uint32_t n = f / NCLS;
      uint32_t c = f - n * NCLS;
      cl = (float)(c + 1u);
      const float* bp = boxes + ((size_t)img * NROI + n) * 4;
      bx0 = bp[0]; bx1 = bp[1]; bx2 = bp[2]; bx3 = bp[3];
      float off = cl * CLS_OFFSET_F;
      a0 = bx0 + off; a1 = bx1 + off; a2 = bx2 + off; a3 = bx3 + off;
    }
  }
  top_adj[slot * 4 + 0] = a0; top_adj[slot * 4 + 1] = a1;
  top_adj[slot * 4 + 2] = a2; top_adj[slot * 4 + 3] = a3;
  top_box[slot * 4 + 0] = bx0; top_box[slot * 4 + 1] = bx1;
  top_box[slot * 4 + 2] = bx2; top_box[slot * 4 + 3] = bx3;
  top_score[slot] = sc;
  top_cls[slot] = cl;
  top_valid[slot] = v;
}

// ---------------------------------------------------------------- kernel 0: zero scratch
__global__ void k_zero(uint32_t* __restrict__ hist, uint32_t* __restrict__ cnt) {
  int t = blockIdx.x * blockDim.x + threadIdx.x;
  if (t < BATCH * NBINS) hist[t] = 0u;
  if (t < BATCH) cnt[t] = 0u;
}

// ---------------------------------------------------------------- kernel 1: box decode
__global__ void k_decode(const float* __restrict__ rois,
                         const float* __restrict__ reg,
                         float* __restrict__ boxes) {
  int t = blockIdx.x * blockDim.x + threadIdx.x;
  if (t >= BATCH * NROI) return;
  const float* r = rois + (size_t)t * 4;
  const float* d = reg + (size_t)t * 4;
  float x1 = r[0], y1 = r[1], x2 = r[2], y2 = r[3];
  float w = x2 - x1, h = y2 - y1;
  float cx = x1 + 0.5f * w, cy = y1 + 0.5f * h;
  float dx = d[0], dy = d[1];
  float dw = fminf(fmaxf(d[2], -MAX_RATIO_F), MAX_RATIO_F);
  float dh = fminf(fmaxf(d[3], -MAX_RATIO_F), MAX_RATIO_F);
  float pw = w * expf(dw), ph = h * expf(dh);
  float pcx = cx + dx * w, pcy = cy + dy * h;
  float* o = boxes + (size_t)t * 4;
  o[0] = pcx - 0.5f * pw;
  o[1] = pcy - 0.5f * ph;
  o[2] = pcx + 0.5f * pw;
  o[3] = pcy + 0.5f * ph;
}

// ---------------------------------------------------------------- kernel 2: score histogram (top 16 bits)
__global__ void k_hist(const float* __restrict__ scores, uint32_t* __restrict__ hist) {
  int t = blockIdx.x * blockDim.x + threadIdx.x;
  if (t >= BATCH * NC) return;
  float s = scores[t];
  if (s > SCORE_THR_F) {
    int img = t / NC;
    uint32_t bin = __float_as_uint(s) >> 16;   // positive floats: uint order == float order
    atomicAdd(&hist[(size_t)img * NBINS + bin], 1u);
  }
}

// ---------------------------------------------------------------- kernel 3: find threshold bin per image
__global__ void k_thresh(const uint32_t* __restrict__ hist, uint32_t* __restrict__ thr) {
  int img = blockIdx.x;
  const uint32_t* h = hist + (size_t)img * NBINS;
  __shared__ uint32_t csum[256];
  int t = threadIdx.x;
  int hi = NBINS - 1 - t * 256;               // descending 256-bin chunks
  uint32_t s = 0;
  for (int b = hi; b > hi - 256; --b) s += h[b];
  csum[t] = s;
  __syncthreads();
  if (t == 0) {
    uint32_t cum = 0;
    int chunk = -1;
    for (int c = 0; c < 256; ++c) {
      if (cum + csum[c] >= (uint32_t)TOPK) { chunk = c; break; }
      cum += csum[c];
    }
    uint32_t tb = 0;
    if (chunk >= 0) {
      int hi2 = NBINS - 1 - chunk * 256;
      for (int b = hi2; b > hi2 - 256; --b) {
        cum += h[b];
        if (cum >= (uint32_t)TOPK) { tb = (uint32_t)b; break; }
      }
    }
    thr[img] = tb;
  }
}

// ---------------------------------------------------------------- kernel 4: candidate compaction
__global__ void k_compact(const float* __restrict__ scores,
                          const uint32_t* __restrict__ thr,
                          uint32_t* __restrict__ cnt,
                          uint64_t* __restrict__ cand) {
  int t = blockIdx.x * blockDim.x + threadIdx.x;
  if (t >= BATCH * NC) return;
  float s = scores[t];
  if (s <= SCORE_THR_F) return;
  int img = t / NC;
  uint32_t f = (uint32_t)(t - img * NC);
  uint32_t bits = __float_as_uint(s);
  if ((bits >> 16) < thr[img]) return;
  uint32_t pos = atomicAdd(&cnt[img], 1u);
  if (pos < CAND_MAX) {
    // key: score bits high, ~index low -> descending sort gives (score desc, index asc)
    cand[(size_t)img * CAND_MAX + pos] = ((uint64_t)bits << 32) | (uint32_t)(~f);
  }
}

// ---------------------------------------------------------------- launcher
extern "C" void kernel_launch(void* const* d_in, const int* in_sizes, int n_in,
                              void* d_out, int out_size, void* d_ws, size_t ws_size,
                              hipStream_t stream) {
  (void)in_sizes; (void)n_in; (void)out_size; (void)ws_size;
  const float* rois = (const float*)d_in[0];   // (B,N,4)
  const float* cls  = (const float*)d_in[1];   // (B*N,C,1,1) == (B, N*C)
  const float* reg  = (const float*)d_in[2];   // (B*N,4,1,1)
  float* out = (float*)d_out;                  // (B, K, 7)
  uint8_t* ws = (uint8_t*)d_ws;

  // workspace layout (bytes)
  const size_t OFF_BOXES = 0;                          // B*N*4 f32      = 1,048,576
  const size_t OFF_HIST  = 1048576;                    // B*65536 u32    = 2,097,152
  const size_t OFF_CNT   = OFF_HIST + 2097152;         // B u32 (pad 256)
  const size_t OFF_THR   = OFF_CNT + 256;              // B u32 (pad 256)
  const size_t OFF_CAND  = OFF_THR + 256;              // B*8192 u64     =   524,288
  const size_t OFF_TADJ  = OFF_CAND + 524288;          // B*1024*4 f32   =   131,072
  const size_t OFF_TBOX  = OFF_TADJ + 131072;          // B*1024*4 f32   =   131,072
  const size_t OFF_TSC   = OFF_TBOX + 131072;          // B*1024 f32     =    32,768
  const size_t OFF_TCL   = OFF_TSC + 32768;            // B*1024 f32     =    32,768
  const size_t OFF_TVA   = OFF_TCL + 32768;            // B*1024 u32     =    32,768
  const size_t OFF_SUP   = OFF_HIST;                   // alias hist (done by then): B*1024*32 u32 = 1,048,576

  float*    boxes     = (float*)(ws + OFF_BOXES);
  uint32_t* hist      = (uint32_t*)(ws + OFF_HIST);
  uint32_t* cnt       = (uint32_t*)(ws + OFF_CNT);
  uint32_t* thr       = (uint32_t*)(ws + OFF_THR);
  uint64_t* cand      = (uint64_t*)(ws + OFF_CAND);
  float*    top_adj   = (float*)(ws + OFF_TADJ);
  float*    top_box   = (float*)(ws + OFF_TBOX);
  float*    top_score = (float*)(ws + OFF_TSC);
  float*    top_cls   = (float*)(ws + OFF_TCL);
  uint32_t* top_valid = (uint32_t*)(ws + OFF_TVA);
  uint32_t* supmask   = (uint32_t*)(ws + OFF_SUP);

  k_zero<<<(BATCH * NBINS + 255) / 256, 256, 0, stream>>>(hist, cnt);
  k_decode<<<(BATCH * NROI + 255) / 256, 256, 0, stream>>>(rois, reg, boxes);
  k_hist<<<(BATCH * NC + 255) / 256, 256, 0, stream>>>(cls, hist);
  k_thresh<<<BATCH, 256, 0, stream>>>(hist, thr);
  k_compact<<<(BATCH * NC + 255) / 256, 256, 0, stream>>>(cls, thr, cnt, cand);
  k_sort<<<BATCH, 1024, 0, stream>>>(cand, cnt, boxes, top_adj, top_box,
                                     top_score, top_cls, top_valid);
  k_iou<<<BATCH * 128, 256, 0, stream>>>(top_adj, supmask);
  k_nms<<<BATCH, 1024, 0, stream>>>(supmask, top_valid, top_box, top_score, top_cls, out);
}